// IEGMN_Layer_6004364280151
// MI455X (gfx1250) — compile-verified
//
#include <hip/hip_runtime.h>
#include <hip/hip_bf16.h>
#include <cstdint>

#define N_NODES 6000
#define NH      ((size_t)N_NODES * 128)

typedef float v2f __attribute__((ext_vector_type(2)));
typedef float v8f __attribute__((ext_vector_type(8)));

// ---------------------------------------------------------------------------
// WMMA helpers (CDNA5 wave32, V_WMMA_F32_16X16X4_F32)
// C/D layout: acc[j] holds row (j + 8*(lane>>4)), col (lane&15)
// A 16x4:  lane l -> row l&15, k-pair k0 + 2*(l>>4)
// B 4x16:  lane l -> col l&15, k-pair k0 + 2*(l>>4)
// ---------------------------------------------------------------------------
__device__ __forceinline__ v8f wmma4(v2f a, v2f b, v8f c) {
  return __builtin_amdgcn_wmma_f32_16x16x4_f32(false, a, false, b, (short)0, c,
                                               false, false);
}

__device__ __forceinline__ v2f ldsA(const float* base, int ld, int k0, int lane) {
  const int m = lane & 15;
  const int k = k0 + ((lane >> 4) << 1);
  const float* p = base + m * ld + k;
  v2f r; r.x = p[0]; r.y = p[1];
  return r;
}

// B fragment from PAIR-PACKED weights: P[k/2][n] = {W[k][n], W[k+1][n]}
// (one aligned b64 load per fragment; k0 is always a multiple of 4)
__device__ __forceinline__ v2f ldBp(const float* __restrict__ P, int k0, int n0,
                                    int lane) {
  const int n = n0 + (lane & 15);
  const int k = k0 + ((lane >> 4) << 1);       // even
  const float* p = P + (size_t)(k >> 1) * 256 + n * 2;
  v2f r; r.x = p[0]; r.y = p[1];
  return r;
}

// B = K^T : B[k][n] = Kb[n*ld + k]  (contiguous float2 per lane already)
__device__ __forceinline__ v2f ldBT(const float* __restrict__ Kb, int ld, int k0,
                                    int lane) {
  const int n = lane & 15;
  const int k = k0 + ((lane >> 4) << 1);
  const float* p = Kb + (size_t)n * ld + k;
  v2f r; r.x = p[0]; r.y = p[1];
  return r;
}

__device__ __forceinline__ float lrelu(float x) { return x >= 0.f ? x : 0.01f * x; }

__device__ __forceinline__ float rowmax16(float v) {
  v = fmaxf(v, __shfl_xor(v, 1));
  v = fmaxf(v, __shfl_xor(v, 2));
  v = fmaxf(v, __shfl_xor(v, 4));
  v = fmaxf(v, __shfl_xor(v, 8));
  return v;
}
__device__ __forceinline__ float rowsum16(float v) {
  v += __shfl_xor(v, 1);
  v += __shfl_xor(v, 2);
  v += __shfl_xor(v, 4);
  v += __shfl_xor(v, 8);
  return v;
}

// LayerNorm over 128 cols of a 16x128 LDS tile (stride 132), 128 threads.
__device__ __forceinline__ void layernorm128(float* buf, const float* __restrict__ g,
                                             const float* __restrict__ b,
                                             float (*stats)[2], int tid) {
  const int r = tid >> 3, seg = tid & 7;
  float sum = 0.f, sq = 0.f;
  #pragma unroll
  for (int i = 0; i < 16; i++) {
    float v = buf[r * 132 + seg * 16 + i];
    sum += v; sq += v * v;
  }
  sum += __shfl_xor(sum, 1); sq += __shfl_xor(sq, 1);
  sum += __shfl_xor(sum, 2); sq += __shfl_xor(sq, 2);
  sum += __shfl_xor(sum, 4); sq += __shfl_xor(sq, 4);
  if (seg == 0) {
    float mean = sum * (1.f / 128.f);
    float var  = sq * (1.f / 128.f) - mean * mean;
    stats[r][0] = mean;
    stats[r][1] = rsqrtf(var + 1e-5f);
  }
  __syncthreads();
  const float mean = stats[r][0], rstd = stats[r][1];
  #pragma unroll
  for (int i = 0; i < 16; i++) {
    int c = seg * 16 + i;
    float v = buf[r * 132 + c];
    buf[r * 132 + c] = g[c] * (v - mean) * rstd + b[c];
  }
}

// 16x128 = (16x128 LDS, stride 132) @ Wp(packed 128x128) + bias -> LDS out
__device__ __forceinline__ void gemm128_lds(const float* in, const float* __restrict__ Wp,
                                            const float* __restrict__ bias, float* outb,
                                            int lane, int wave, int act) {
  const int hi = lane >> 4, cn = lane & 15;
  const int n0 = wave * 32;
  v8f a0 = {}, a1 = {};
  for (int k0 = 0; k0 < 128; k0 += 4) {
    v2f a  = ldsA(in, 132, k0, lane);
    v2f b0 = ldBp(Wp, k0, n0, lane);
    v2f b1 = ldBp(Wp, k0, n0 + 16, lane);
    a0 = wmma4(a, b0, a0);
    a1 = wmma4(a, b1, a1);
  }
  #pragma unroll
  for (int j = 0; j < 8; j++) {
    int r = j + 8 * hi;
    float x0 = a0[j] + bias[n0 + cn];
    float x1 = a1[j] + bias[n0 + 16 + cn];
    if (act) { x0 = lrelu(x0); x1 = lrelu(x1); }
    outb[r * 132 + n0 + cn] = x0;
    outb[r * 132 + n0 + 16 + cn] = x1;
  }
}

// ---------------------------------------------------------------------------
// Pack W[K x 128] row-major into pair-interleaved P: P[k2*256 + n*2 + {0,1}]
// = {W[2k2][n], W[2k2+1][n]}, zero-filled past K.
// ---------------------------------------------------------------------------
__global__ void pack_pairs(const float* __restrict__ W, float* __restrict__ P,
                           int K, int pairs) {
  int i = blockIdx.x * blockDim.x + threadIdx.x;
  if (i >= pairs * 128) return;
  int k2 = i >> 7, n = i & 127;
  int k = 2 * k2;
  float a = (k < K)     ? W[(size_t)k * 128 + n]       : 0.f;
  float b = (k + 1 < K) ? W[(size_t)(k + 1) * 128 + n] : 0.f;
  P[(size_t)k2 * 256 + n * 2]     = a;
  P[(size_t)k2 * 256 + n * 2 + 1] = b;
}

// ---------------------------------------------------------------------------
// Projection GEMM: C = act(A[6000x128] @ W[128x128]); act=1 -> lrelu.
// pack=1 writes C in pair-packed (B-operand) layout for the attention PV GEMM.
// ---------------------------------------------------------------------------
__global__ void __launch_bounds__(128) proj_gemm(const float* __restrict__ A,
                                                 const float* __restrict__ Wp,
                                                 float* __restrict__ C, int act,
                                                 int pack) {
  __shared__ float at[16 * 132];
  const int tid = threadIdx.x, lane = tid & 31, wave = tid >> 5;
  const int row0 = blockIdx.x * 16;
  for (int i = tid; i < 16 * 128; i += 128) {
    int r = i >> 7, c = i & 127;
    at[r * 132 + c] = A[(size_t)(row0 + r) * 128 + c];
  }
  __syncthreads();
  const int n0 = wave * 32;
  v8f a0 = {}, a1 = {};
  for (int k0 = 0; k0 < 128; k0 += 4) {
    v2f a  = ldsA(at, 132, k0, lane);
    v2f b0 = ldBp(Wp, k0, n0, lane);
    v2f b1 = ldBp(Wp, k0, n0 + 16, lane);
    a0 = wmma4(a, b0, a0);
    a1 = wmma4(a, b1, a1);
  }
  const int hi = lane >> 4, cn = lane & 15;
  #pragma unroll
  for (int j = 0; j < 8; j++) {
    int r = row0 + j + 8 * hi;
    float x0 = a0[j], x1 = a1[j];
    if (act) { x0 = lrelu(x0); x1 = lrelu(x1); }
    if (pack) {
      C[(size_t)(r >> 1) * 256 + (n0 + cn) * 2 + (r & 1)]      = x0;
      C[(size_t)(r >> 1) * 256 + (n0 + 16 + cn) * 2 + (r & 1)] = x1;
    } else {
      C[(size_t)r * 128 + n0 + cn] = x0;
      C[(size_t)r * 128 + n0 + 16 + cn] = x1;
    }
  }
}

// ---------------------------------------------------------------------------
// Flash-style masked cross attention:
// out = softmax(mask*(Q K^T) - 1000*(1-mask), axis=1) @ V
// 256 thr / block; block = 16 query rows; 8 waves split key chunks.
// V is pair-packed (B-operand layout).
// ---------------------------------------------------------------------------
__global__ void __launch_bounds__(256) attn_kernel(const float* __restrict__ Q,
                                                   const float* __restrict__ K,
                                                   const float* __restrict__ Vp,
                                                   const float* __restrict__ mask,
                                                   int tmask, float* __restrict__ out) {
  __shared__ float qt[16 * 132];
  __shared__ float pbuf[8][16 * 18];
  __shared__ float outbuf[16 * 128];
  __shared__ float mbuf[8][16], lbuf[8][16];
  __shared__ float Mrow[16], Lrow[16];
  const int tid = threadIdx.x, lane = tid & 31, wave = tid >> 5;
  const int hi = lane >> 4, cn = lane & 15;
  const int qrow0 = blockIdx.x * 16;

  for (int i = tid; i < 2048; i += 256) {
    int r = i >> 7, c = i & 127;
    qt[r * 132 + c] = Q[(size_t)(qrow0 + r) * 128 + c];
  }
  __syncthreads();

  float mr[8], lr[8];
  v8f acc[8];
  #pragma unroll
  for (int j = 0; j < 8; j++) { mr[j] = -3.0e38f; lr[j] = 0.f; acc[j] = (v8f){}; }
  float* pw = &pbuf[wave][0];

  for (int ch = wave; ch < N_NODES / 16; ch += 8) {
    const int key0 = ch * 16;
    // S = Q @ K^T (16x16 tile)
    v8f s = {};
    for (int k0 = 0; k0 < 128; k0 += 4) {
      v2f a = ldsA(qt, 132, k0, lane);
      v2f b = ldBT(K + (size_t)key0 * 128, 128, k0, lane);
      s = wmma4(a, b, s);
    }
    const int col = key0 + cn;
    float sc[8];
    #pragma unroll
    for (int j = 0; j < 8; j++) {
      const int row = qrow0 + j + 8 * hi;
      float mv = tmask ? mask[(size_t)col * N_NODES + row]
                       : mask[(size_t)row * N_NODES + col];
      float sj = mv * s[j] - 1000.f * (1.f - mv);
      float mnew = fmaxf(mr[j], rowmax16(sj));
      sc[j] = __expf(mr[j] - mnew);
      float pj = __expf(sj - mnew);
      lr[j] = lr[j] * sc[j] + rowsum16(pj);
      mr[j] = mnew;
      pw[(j + 8 * hi) * 18 + cn] = pj;   // stash P for transpose to A-frag
    }
    #pragma unroll
    for (int t = 0; t < 8; t++)
      #pragma unroll
      for (int j = 0; j < 8; j++) acc[t][j] *= sc[j];
    asm volatile("s_wait_dscnt 0" ::: "memory");  // wave-local P visible
    #pragma unroll
    for (int k0 = 0; k0 < 16; k0 += 4) {
      v2f a = ldsA(pw, 18, k0, lane);
      #pragma unroll
      for (int t = 0; t < 8; t++) {
        v2f b = ldBp(Vp, key0 + k0, t * 16, lane);  // one b64 per fragment
        acc[t] = wmma4(a, b, acc[t]);
      }
    }
  }

  if (cn == 0) {
    #pragma unroll
    for (int j = 0; j < 8; j++) {
      mbuf[wave][j + 8 * hi] = mr[j];
      lbuf[wave][j + 8 * hi] = lr[j];
    }
  }
  __syncthreads();
  for (int i = tid; i < 2048; i += 256) outbuf[i] = 0.f;
  if (tid < 16) {
    float M = -3.0e38f;
    for (int w = 0; w < 8; w++) M = fmaxf(M, mbuf[w][tid]);
    float L = 0.f;
    for (int w = 0; w < 8; w++) L += lbuf[w][tid] * __expf(mbuf[w][tid] - M);
    Mrow[tid] = M;
    Lrow[tid] = L > 0.f ? 1.f / L : 0.f;
  }
  __syncthreads();
  #pragma unroll
  for (int j = 0; j < 8; j++) {
    float e = __expf(mr[j] - Mrow[j + 8 * hi]);
    #pragma unroll
    for (int t = 0; t < 8; t++)
      atomicAdd(&outbuf[(j + 8 * hi) * 128 + t * 16 + cn], acc[t][j] * e);
  }
  __syncthreads();
  for (int i = tid; i < 2048; i += 256) {
    int r = i >> 7, c = i & 127;
    out[(size_t)(qrow0 + r) * 128 + c] = outbuf[i] * Lrow[r];
  }
}

// ---------------------------------------------------------------------------
// Edge kernel: 16 edges/block, fused edge-MLP + coors-MLP + atomic scatter.
// eW1p is packed/zero-padded to 272 rows -> uniform K loop, no tail.
// ---------------------------------------------------------------------------
#define CAT_LD 274
__global__ void __launch_bounds__(128) edge_kernel(
    const float* __restrict__ coors, const float* __restrict__ h,
    const int* __restrict__ src, const int* __restrict__ dst,
    const float* __restrict__ eW1p, const float* __restrict__ eb1,
    const float* __restrict__ eg1, const float* __restrict__ ebt1,
    const float* __restrict__ eW2p, const float* __restrict__ eb2,
    const float* __restrict__ cW1p, const float* __restrict__ cb1,
    const float* __restrict__ cg1, const float* __restrict__ cbt1,
    const float* __restrict__ cW2, const float* __restrict__ cb2,
    float* __restrict__ sum_msg, float* __restrict__ sum_x,
    float* __restrict__ cnt, int nE) {
  __shared__ float cat[16 * CAT_LD];
  __shared__ float buf1[16 * 132];
  __shared__ float buf2[16 * 132];
  __shared__ float stats[16][2];
  __shared__ float xrel[16][3];
  __shared__ float coefs[16];
  __shared__ int sidx[16], didx[16];
  const int tid = threadIdx.x, lane = tid & 31, wave = tid >> 5;
  const int hi = lane >> 4, cn = lane & 15;
  const int e0 = blockIdx.x * 16;

  if (tid < 16) {
    int e = e0 + tid;
    if (e < nE) {
      int s = src[e], d = dst[e];
      sidx[tid] = s; didx[tid] = d;
      float x0 = coors[s * 3 + 0] - coors[d * 3 + 0];
      float x1 = coors[s * 3 + 1] - coors[d * 3 + 1];
      float x2 = coors[s * 3 + 2] - coors[d * 3 + 2];
      xrel[tid][0] = x0; xrel[tid][1] = x1; xrel[tid][2] = x2;
      float d2 = x0 * x0 + x1 * x1 + x2 * x2;
      float sig = 1.f;
      for (int j = 0; j < 15; j++) {
        cat[tid * CAT_LD + 256 + j] = __expf(-d2 / sig);
        sig *= 1.5f;
      }
      cat[tid * CAT_LD + 271] = 0.f;  // K padding column
    } else {
      sidx[tid] = 0; didx[tid] = -1;
      xrel[tid][0] = xrel[tid][1] = xrel[tid][2] = 0.f;
      for (int j = 0; j < 16; j++) cat[tid * CAT_LD + 256 + j] = 0.f;
    }
  }
  __syncthreads();
  // gather h[src] | h[dst] -> cat[:,0:256]
  for (int i = tid; i < 16 * 256; i += 128) {
    int e = i >> 8, c = i & 255;
    int node = (didx[e] >= 0) ? ((c < 128) ? sidx[e] : didx[e]) : 0;
    cat[e * CAT_LD + c] = h[(size_t)node * 128 + (c & 127)];
  }
  __syncthreads();

  // GEMM1: [16x272] @ eW1p(272x128 packed) -> lrelu -> buf1
  {
    const int n0 = wave * 32;
    v8f a0 = {}, a1 = {};
    for (int k0 = 0; k0 < 272; k0 += 4) {
      v2f a  = ldsA(cat, CAT_LD, k0, lane);
      v2f b0 = ldBp(eW1p, k0, n0, lane);
      v2f b1 = ldBp(eW1p, k0, n0 + 16, lane);
      a0 = wmma4(a, b0, a0);
      a1 = wmma4(a, b1, a1);
    }
    #pragma unroll
    for (int j = 0; j < 8; j++) {
      int r = j + 8 * hi;
      buf1[r * 132 + n0 + cn]      = lrelu(a0[j] + eb1[n0 + cn]);
      buf1[r * 132 + n0 + 16 + cn] = lrelu(a1[j] + eb1[n0 + 16 + cn]);
    }
  }
  __syncthreads();
  layernorm128(buf1, eg1, ebt1, stats, tid);
  __syncthreads();
  gemm128_lds(buf1, eW2p, eb2, buf2, lane, wave, 0);  // msg -> buf2
  __syncthreads();
  gemm128_lds(buf2, cW1p, cb1, buf1, lane, wave, 1);  // lrelu(msg@cW1+cb1)
  __syncthreads();
  layernorm128(buf1, cg1, cbt1, stats, tid);
  __syncthreads();
  // coef = buf1 @ cW2[128x1] + cb2
  {
    int r = tid >> 3, seg = tid & 7;
    float sum = 0.f;
    #pragma unroll
    for (int i = 0; i < 16; i++) {
      int c = seg * 16 + i;
      sum += buf1[r * 132 + c] * cW2[c];
    }
    sum += __shfl_xor(sum, 1);
    sum += __shfl_xor(sum, 2);
    sum += __shfl_xor(sum, 4);
    if (seg == 0) coefs[r] = sum + cb2[0];
  }
  __syncthreads();
  // scatter: sum_msg[dst] += msg ; sum_x[dst] += x_rel*coef ; cnt[dst] += 1
  for (int i = tid; i < 2048; i += 128) {
    int e = i >> 7, c = i & 127;
    int d = didx[e];
    if (d >= 0) atomicAdd(&sum_msg[(size_t)d * 128 + c], buf2[e * 132 + c]);
  }
  if (tid < 48) {
    int e = tid / 3, ax = tid % 3;
    int d = didx[e];
    if (d >= 0) atomicAdd(&sum_x[d * 3 + ax], xrel[e][ax] * coefs[e]);
  }
  if (tid < 16) {
    int d = didx[tid];
    if (d >= 0) atomicAdd(&cnt[d], 1.f);
  }
}

// ---------------------------------------------------------------------------
// Node kernel: in=[h|am|cm|orig] (448) -> MLP -> 0.5*mlp + 0.5*h
// ---------------------------------------------------------------------------
#define NCAT_LD 450
__global__ void __launch_bounds__(128) node_kernel(
    const float* __restrict__ h, const float* __restrict__ summ,
    const float* __restrict__ cnt, const float* __restrict__ cm,
    const float* __restrict__ orig,
    const float* __restrict__ nW1p, const float* __restrict__ nb1,
    const float* __restrict__ ng1, const float* __restrict__ nbt1,
    const float* __restrict__ nW2p, const float* __restrict__ nb2,
    float* __restrict__ hout) {
  __shared__ float cat[16 * NCAT_LD];
  __shared__ float buf1[16 * 132];
  __shared__ float stats[16][2];
  const int tid = threadIdx.x, lane = tid & 31, wave = tid >> 5;
  const int hi = lane >> 4, cn = lane & 15;
  const int node0 = blockIdx.x * 16;

  for (int i = tid; i < 2048; i += 128) {
    int e = i >> 7, c = i & 127;
    int node = node0 + e;
    cat[e * NCAT_LD + c] = h[(size_t)node * 128 + c];
    float ct = cnt[node];
    cat[e * NCAT_LD + 128 + c] = ct > 0.f ? summ[(size_t)node * 128 + c] / ct : 0.f;
    cat[e * NCAT_LD + 256 + c] = cm[(size_t)node * 128 + c];
  }
  for (int i = tid; i < 1024; i += 128) {
    int e = i >> 6, c = i & 63;
    cat[e * NCAT_LD + 384 + c] = orig[(size_t)(node0 + e) * 64 + c];
  }
  __syncthreads();
  // GEMM1: 448 -> 128, lrelu
  {
    const int n0 = wave * 32;
    v8f a0 = {}, a1 = {};
    for (int k0 = 0; k0 < 448; k0 += 4) {
      v2f a  = ldsA(cat, NCAT_LD, k0, lane);
      v2f b0 = ldBp(nW1p, k0, n0, lane);
      v2f b1 = ldBp(nW1p, k0, n0 + 16, lane);
      a0 = wmma4(a, b0, a0);
      a1 = wmma4(a, b1, a1);
    }
    #pragma unroll
    for (int j = 0; j < 8; j++) {
      int r = j + 8 * hi;
      buf1[r * 132 + n0 + cn]      = lrelu(a0[j] + nb1[n0 + cn]);
      buf1[r * 132 + n0 + 16 + cn] = lrelu(a1[j] + nb1[n0 + 16 + cn]);
    }
  }
  __syncthreads();
  layernorm128(buf1, ng1, nbt1, stats, tid);
  __syncthreads();
  // GEMM2 + skip
  {
    const int n0 = wave * 32;
    v8f a0 = {}, a1 = {};
    for (int k0 = 0; k0 < 128; k0 += 4) {
      v2f a  = ldsA(buf1, 132, k0, lane);
      v2f b0 = ldBp(nW2p, k0, n0, lane);
      v2f b1 = ldBp(nW2p, k0, n0 + 16, lane);
      a0 = wmma4(a, b0, a0);
      a1 = wmma4(a, b1, a1);
    }
    #pragma unroll
    for (int j = 0; j < 8; j++) {
      int r = j + 8 * hi;
      size_t g0 = (size_t)(node0 + r) * 128 + n0 + cn;
      size_t g1 = g0 + 16;
      hout[g0] = 0.5f * (a0[j] + nb2[n0 + cn]) + 0.5f * h[g0];
      hout[g1] = 0.5f * (a1[j] + nb2[n0 + 16 + cn]) + 0.5f * h[g1];
    }
  }
}

// x_final = 0.2*orig_coors + 0.8*coors + seg_mean(x_rel*coef)
__global__ void coords_kernel(const float* __restrict__ oc, const float* __restrict__ c,
                              const float* __restrict__ sx, const float* __restrict__ cnt,
                              float* __restrict__ out, int n) {
  int i = blockIdx.x * blockDim.x + threadIdx.x;
  if (i >= n * 3) return;
  int node = i / 3;
  float ct = cnt[node];
  float xu = ct > 0.f ? sx[i] / ct : 0.f;
  out[i] = 0.2f * oc[i] + 0.8f * c[i] + xu;
}

// ---------------------------------------------------------------------------
extern "C" void kernel_launch(void* const* d_in, const int* in_sizes, int n_in,
                              void* d_out, int out_size, void* d_ws, size_t ws_size,
                              hipStream_t stream) {
  const float* coors_lig = (const float*)d_in[0];
  const float* h_lig     = (const float*)d_in[1];
  const float* orig_lig  = (const float*)d_in[2];
  const float* oc_lig    = (const float*)d_in[3];
  const float* coors_rec = (const float*)d_in[4];
  const float* h_rec     = (const float*)d_in[5];
  const float* orig_rec  = (const float*)d_in[6];
  const float* oc_rec    = (const float*)d_in[7];
  const int* lig_src = (const int*)d_in[8];
  const int* lig_dst = (const int*)d_in[9];
  const int* rec_src = (const int*)d_in[10];
  const int* rec_dst = (const int*)d_in[11];
  const float* mask = (const float*)d_in[12];
  const float* eW1 = (const float*)d_in[13];
  const float* eb1 = (const float*)d_in[14];
  const float* eg1 = (const float*)d_in[15];
  const float* ebt1 = (const float*)d_in[16];
  const float* eW2 = (const float*)d_in[17];
  const float* eb2 = (const float*)d_in[18];
  const float* qW = (const float*)d_in[19];
  const float* kW = (const float*)d_in[20];
  const float* vW = (const float*)d_in[21];
  const float* nW1 = (const float*)d_in[22];
  const float* nb1 = (const float*)d_in[23];
  const float* ng1 = (const float*)d_in[24];
  const float* nbt1 = (const float*)d_in[25];
  const float* nW2 = (const float*)d_in[26];
  const float* nb2 = (const float*)d_in[27];
  const float* cW1 = (const float*)d_in[28];
  const float* cb1 = (const float*)d_in[29];
  const float* cg1 = (const float*)d_in[30];
  const float* cbt1 = (const float*)d_in[31];
  const float* cW2 = (const float*)d_in[32];
  const float* cb2 = (const float*)d_in[33];
  const int nE = in_sizes[8];

  float* ws = (float*)d_ws;
  float* q_l = ws + 0 * NH;
  float* k_l = ws + 1 * NH;
  float* v_l = ws + 2 * NH;   // pair-packed
  float* q_r = ws + 3 * NH;
  float* k_r = ws + 4 * NH;
  float* v_r = ws + 5 * NH;   // pair-packed
  float* cm_l = ws + 6 * NH;
  float* cm_r = ws + 7 * NH;
  float* sm_l = ws + 8 * NH;
  float* sm_r = ws + 9 * NH;
  float* sx_l = ws + 10 * NH;
  float* sx_r = sx_l + 3 * N_NODES;
  float* cnt_l = sx_r + 3 * N_NODES;
  float* cnt_r = cnt_l + N_NODES;
  // pair-packed weights
  float* eW1p = cnt_r + N_NODES;        // 136*256
  float* eW2p = eW1p + 136 * 256;       // 64*256
  float* cW1p = eW2p + 64 * 256;        // 64*256
  float* nW1p = cW1p + 64 * 256;        // 224*256
  float* nW2p = nW1p + 224 * 256;       // 64*256
  float* qWp  = nW2p + 64 * 256;        // 64*256
  float* kWp  = qWp + 64 * 256;         // 64*256
  float* vWp  = kWp + 64 * 256;         // 64*256

  // zero the atomic accumulators (contiguous region: sm_l .. cnt_r)
  size_t accum_floats = 2 * NH + 8 * (size_t)N_NODES;
  hipMemsetAsync(sm_l, 0, accum_floats * sizeof(float), stream);

  // pack weights into pair-interleaved B layout
  pack_pairs<<<(136 * 128 + 255) / 256, 256, 0, stream>>>(eW1, eW1p, 271, 136);
  pack_pairs<<<(64 * 128 + 255) / 256, 256, 0, stream>>>(eW2, eW2p, 128, 64);
  pack_pairs<<<(64 * 128 + 255) / 256, 256, 0, stream>>>(cW1, cW1p, 128, 64);
  pack_pairs<<<(224 * 128 + 255) / 256, 256, 0, stream>>>(nW1, nW1p, 448, 224);
  pack_pairs<<<(64 * 128 + 255) / 256, 256, 0, stream>>>(nW2, nW2p, 128, 64);
  pack_pairs<<<(64 * 128 + 255) / 256, 256, 0, stream>>>(qW, qWp, 128, 64);
  pack_pairs<<<(64 * 128 + 255) / 256, 256, 0, stream>>>(kW, kWp, 128, 64);
  pack_pairs<<<(64 * 128 + 255) / 256, 256, 0, stream>>>(vW, vWp, 128, 64);

  const int ntiles = N_NODES / 16;  // 375
  proj_gemm<<<ntiles, 128, 0, stream>>>(h_lig, qWp, q_l, 1, 0);
  proj_gemm<<<ntiles, 128, 0, stream>>>(h_lig, kWp, k_l, 1, 0);
  proj_gemm<<<ntiles, 128, 0, stream>>>(h_lig, vWp, v_l, 0, 1);
  proj_gemm<<<ntiles, 128, 0, stream>>>(h_rec, qWp, q_r, 1, 0);
  proj_gemm<<<ntiles, 128, 0, stream>>>(h_rec, kWp, k_r, 1, 0);
  proj_gemm<<<ntiles, 128, 0, stream>>>(h_rec, vWp, v_r, 0, 1);

  attn_kernel<<<ntiles, 256, 0, stream>>>(q_l, k_r, v_r, mask, 0, cm_l);
  attn_kernel<<<ntiles, 256, 0, stream>>>(q_r, k_l, v_l, mask, 1, cm_r);

  int eblocks = (nE + 15) / 16;
  edge_kernel<<<eblocks, 128, 0, stream>>>(coors_lig, h_lig, lig_src, lig_dst,
      eW1p, eb1, eg1, ebt1, eW2p, eb2, cW1p, cb1, cg1, cbt1, cW2, cb2,
      sm_l, sx_l, cnt_l, nE);
  edge_kernel<<<eblocks, 128, 0, stream>>>(coors_rec, h_rec, rec_src, rec_dst,
      eW1p, eb1, eg1, ebt1, eW2p, eb2, cW1p, cb1, cg1, cbt1, cW2, cb2,
      sm_r, sx_r, cnt_r, nE);

  float* out_xl = (float*)d_out;
  float* out_hl = out_xl + 3 * N_NODES;
  float* out_xr = out_hl + NH;
  float* out_hr = out_xr + 3 * N_NODES;

  node_kernel<<<ntiles, 128, 0, stream>>>(h_lig, sm_l, cnt_l, cm_l, orig_lig,
      nW1p, nb1, ng1, nbt1, nW2p, nb2, out_hl);
  node_kernel<<<ntiles, 128, 0, stream>>>(h_rec, sm_r, cnt_r, cm_r, orig_rec,
      nW1p, nb1, ng1, nbt1, nW2p, nb2, out_hr);

  int cthreads = 3 * N_NODES;
  coords_kernel<<<(cthreads + 255) / 256, 256, 0, stream>>>(oc_lig, coors_lig, sx_l,
                                                            cnt_l, out_xl, N_NODES);
  coords_kernel<<<(cthreads + 255) / 256, 256, 0, stream>>>(oc_rec, coors_rec, sx_r,
                                                            cnt_r, out_xr, N_NODES);
}